// WinMSA_39633958207522
// MI455X (gfx1250) — compile-verified
//
#include <hip/hip_runtime.h>

typedef __attribute__((ext_vector_type(2))) float v2f;
typedef __attribute__((ext_vector_type(8))) float v8f;

#define B_   32
#define WN_  64
#define S_   49
#define D_   384
#define H_   12
#define DH_  32
#define R_   97          // 2*S-1
#define BW_  (B_ * WN_)  // 2048

// LDS strides (floats), padded so 16 consecutive rows map to 16 distinct banks.
#define QKV_STR 34       // Q/K/V tiles: 64 rows x 32 cols
#define S_STR   66       // score/probability tile: 64 rows x 64 cols (keys)
#define SC_STR  98       // pe-score tile: 49 rows x 97 bins

#define K2_ROWS 32       // projection kernel: rows staged per block
#define K2_STR  388      // stride for staged A tile (row*388*4 is 16B aligned)

__device__ __forceinline__ v8f wmma_f32(v2f a, v2f b, v8f c) {
  // D = A(16x4) * B(4x16) + C, fp32 in / fp32 acc  -> v_wmma_f32_16x16x4_f32
  return __builtin_amdgcn_wmma_f32_16x16x4_f32(
      /*neg_a=*/false, a, /*neg_b=*/false, b,
      /*c_mod=*/(short)0, c, /*reuse_a=*/false, /*reuse_b=*/false);
}

// ---------------------------------------------------------------------------
// Kernel 1: one block per (window, head).
//   QKV projection -> LDS, q@pe^T -> LDS, q@k^T*scale + Toeplitz(rpe) -> LDS,
//   softmax, p@v -> workspace att[BW][S][D] (head slice h*DH..h*DH+31).
// ---------------------------------------------------------------------------
__global__ __launch_bounds__(256)
void winmsa_attn_kernel(const float* __restrict__ X,
                        const float* __restrict__ Wqkv,
                        const float* __restrict__ pe,
                        float* __restrict__ att) {
  __shared__ float Qs[64 * QKV_STR];
  __shared__ float Ks[64 * QKV_STR];
  __shared__ float Vs[64 * QKV_STR];
  __shared__ float Ss[64 * S_STR];
  __shared__ float SCs[S_ * SC_STR];

  const int bw   = blockIdx.x / H_;
  const int h    = blockIdx.x % H_;
  const int tid  = threadIdx.x;
  const int wave = tid >> 5;
  const int lane = tid & 31;
  const int lrow = lane & 15;   // tile row (A) / tile col (B, C/D)
  const int lhi  = lane >> 4;   // 0 or 1
  const int koff = lhi << 1;    // K sub-offset within a x4 step

  const float* __restrict__ Xw = X + (size_t)bw * S_ * D_;

  // ---- Phase 1: [49x384] @ Wslice^T -> Q|K|V (96 cols), rows 49..63 zeroed.
  // K-loop batched: 16 loads in flight, then 8 back-to-back WMMAs.
  for (int t = wave; t < 24; t += 8) {          // 4 M-tiles x 6 N-tiles
    const int mi = t & 3;
    const int ni = t >> 2;                      // 0,1 -> Q ; 2,3 -> K ; 4,5 -> V
    const int m0 = mi * 16;
    const int arow   = m0 + lrow;
    const bool avalid = (arow < S_);
    const float* __restrict__ Arow = Xw + (size_t)(avalid ? arow : 0) * D_;
    const int wrow = (ni >> 1) * D_ + h * DH_ + (ni & 1) * 16 + lrow;
    const float* __restrict__ Brow = Wqkv + (size_t)wrow * D_;
    __builtin_prefetch(Brow, 0, 0);
    v8f c = {};
    for (int k = 0; k < D_; k += 32) {          // 8 WMMA steps per batch
      v2f a[8], b[8];
#pragma unroll
      for (int u = 0; u < 8; ++u) {
        a[u] = *(const v2f*)(Arow + k + 4 * u + koff);
        b[u] = *(const v2f*)(Brow + k + 4 * u + koff);
      }
      if (!avalid) {
#pragma unroll
        for (int u = 0; u < 8; ++u) { a[u].x = 0.0f; a[u].y = 0.0f; }
      }
#pragma unroll
      for (int u = 0; u < 8; ++u) c = wmma_f32(a[u], b[u], c);
    }
    float* dst = (ni >> 1) == 0 ? Qs : ((ni >> 1) == 1 ? Ks : Vs);
    const int cb = (ni & 1) * 16 + lrow;
#pragma unroll
    for (int j = 0; j < 8; ++j) {
      const int m = m0 + j + (lhi << 3);
      dst[m * QKV_STR + cb] = c[j];
    }
  }
  __syncthreads();

  // ---- Phase 2: sc = Q @ pe_h^T  ([49 x 97], K = 32)
  for (int t = wave; t < 28; t += 8) {          // 4 M-tiles x 7 N-tiles (112 cols)
    const int mi = t & 3;
    const int ni = t >> 2;
    const int m0 = mi * 16;
    int r = ni * 16 + lrow;
    if (r > R_ - 1) r = R_ - 1;                 // clamp (cols >= 97 unused)
    const float* __restrict__ Brow = pe + (size_t)r * D_ + h * DH_;
    v8f c = {};
    {
      v2f a[8], b[8];
#pragma unroll
      for (int u = 0; u < 8; ++u) {
        a[u] = *(const v2f*)(&Qs[(m0 + lrow) * QKV_STR + 4 * u + koff]);
        b[u] = *(const v2f*)(Brow + 4 * u + koff);
      }
#pragma unroll
      for (int u = 0; u < 8; ++u) c = wmma_f32(a[u], b[u], c);
    }
#pragma unroll
    for (int j = 0; j < 8; ++j) {
      const int m = m0 + j + (lhi << 3);
      const int n = ni * 16 + lrow;
      if (m < S_ && n < R_) SCs[m * SC_STR + n] = c[j];
    }
  }
  __syncthreads();

  // ---- Phase 3: attn = (Q @ K^T) * scale + rpe gather  -> Ss (keys>=49 -> 0)
  const float scale = 0.17677669529663687f;     // DH^-0.5 = 1/sqrt(32)
  for (int t = wave * 2; t < wave * 2 + 2; ++t) {  // 16 tiles, 2 per wave
    const int mi = t & 3;
    const int ni = t >> 2;
    const int m0 = mi * 16;
    const int n0 = ni * 16;
    v8f c = {};
    {
      v2f a[8], b[8];
#pragma unroll
      for (int u = 0; u < 8; ++u) {
        a[u] = *(const v2f*)(&Qs[(m0 + lrow) * QKV_STR + 4 * u + koff]);
        b[u] = *(const v2f*)(&Ks[(n0 + lrow) * QKV_STR + 4 * u + koff]);
      }
#pragma unroll
      for (int u = 0; u < 8; ++u) c = wmma_f32(a[u], b[u], c);
    }
#pragma unroll
    for (int j = 0; j < 8; ++j) {
      const int q   = m0 + j + (lhi << 3);
      const int key = n0 + lrow;
      float val = 0.0f;                          // pad keys -> exp skipped
      if (key < S_ && q < S_) {
        const int g = key - q + (S_ - 1);        // Toeplitz bin, in [0, 96]
        val = c[j] * scale + SCs[q * SC_STR + g];
      } else if (key < S_) {
        val = c[j] * scale;                      // dead rows: keep finite
      }
      Ss[q * S_STR + key] = val;
    }
  }
  __syncthreads();

  // ---- Phase 4: row softmax over keys < 49 (cols 49..63 stay 0)
  if (tid < 64) {
    float* row = &Ss[tid * S_STR];
    float m = -1e30f;
    for (int j = 0; j < S_; ++j) m = fmaxf(m, row[j]);
    float sum = 0.0f;
    for (int j = 0; j < S_; ++j) { const float e = __expf(row[j] - m); row[j] = e; sum += e; }
    const float inv = 1.0f / sum;
    for (int j = 0; j < S_; ++j) row[j] *= inv;
  }
  __syncthreads();

  // ---- Phase 5: out_h = P @ V  ([49 x 32], K = 64 with zero tail)
  {
    const int mi = wave & 3;
    const int ni = wave >> 2;                    // 8 tiles, 1 per wave
    const int m0 = mi * 16;
    v8f c = {};
#pragma unroll
    for (int kb = 0; kb < 64; kb += 32) {
      v2f a[8], b[8];
#pragma unroll
      for (int u = 0; u < 8; ++u) {
        const int k = kb + 4 * u;
        a[u] = *(const v2f*)(&Ss[(m0 + lrow) * S_STR + k + koff]);
        const int key = k + koff;
        b[u].x = Vs[key * QKV_STR + ni * 16 + lrow];
        b[u].y = Vs[(key + 1) * QKV_STR + ni * 16 + lrow];
      }
#pragma unroll
      for (int u = 0; u < 8; ++u) c = wmma_f32(a[u], b[u], c);
    }
#pragma unroll
    for (int j = 0; j < 8; ++j) {
      const int q = m0 + j + (lhi << 3);
      if (q < S_)
        att[((size_t)bw * S_ + q) * D_ + h * DH_ + ni * 16 + lrow] = c[j];
    }
  }
}

// ---------------------------------------------------------------------------
// Kernel 2: out = att[100352 x 384] @ W_o^T + b_o.  One block = 32 rows x 384.
// A tile staged into LDS with GLOBAL_LOAD_ASYNC_TO_LDS_B128 (ASYNCcnt path),
// then reused by all 24 N-tiles from LDS.
// ---------------------------------------------------------------------------
__global__ __launch_bounds__(256)
void winmsa_proj_kernel(const float* __restrict__ att,
                        const float* __restrict__ Wo,
                        const float* __restrict__ bo,
                        float* __restrict__ out) {
  __shared__ float As[K2_ROWS * K2_STR];        // 32 x 384 (+pad) ~= 49.7 KB

  const int tid  = threadIdx.x;
  const int wave = tid >> 5;
  const int lane = tid & 31;
  const int lrow = lane & 15;
  const int lhi  = lane >> 4;
  const int koff = lhi << 1;
  const size_t m0blk = (size_t)blockIdx.x * K2_ROWS;

  // ---- Async stage: 32 rows x 96 float4 = 3072 B128 transfers, 12 per lane.
#pragma unroll
  for (int it = 0; it < 12; ++it) {
    const int idx = tid + it * 256;             // 0..3071
    const int row = idx / 96;
    const int c4  = idx % 96;
    const float* gsrc = att + (m0blk + (size_t)row) * D_ + c4 * 4;
    const unsigned ldsoff =
        (unsigned)(uintptr_t)&As[row * K2_STR + c4 * 4];  // low 32 bits = LDS offset
    asm volatile("global_load_async_to_lds_b128 %0, %1, off"
                 :: "v"(ldsoff), "v"(gsrc)
                 : "memory");
  }
  asm volatile("s_wait_asynccnt 0x0" ::: "memory");
  __syncthreads();

  // ---- 2 M-tiles x 24 N-tiles, 6 tiles per wave; K batched 8 WMMAs deep.
  for (int t = wave; t < 48; t += 8) {
    const int mi = t & 1;
    const int ni = t >> 1;
    const int m0 = mi * 16;
    const float* Arow = &As[(m0 + lrow) * K2_STR];
    const float* __restrict__ Brow = Wo + (size_t)(ni * 16 + lrow) * D_;
    __builtin_prefetch(Brow, 0, 0);
    v8f c = {};
    for (int k = 0; k < D_; k += 32) {
      v2f a[8], b[8];
#pragma unroll
      for (int u = 0; u < 8; ++u) {
        a[u] = *(const v2f*)(Arow + k + 4 * u + koff);
        b[u] = *(const v2f*)(Brow + k + 4 * u + koff);
      }
#pragma unroll
      for (int u = 0; u < 8; ++u) c = wmma_f32(a[u], b[u], c);
    }
    const float bias = bo[ni * 16 + lrow];
#pragma unroll
    for (int j = 0; j < 8; ++j) {
      const size_t m = m0blk + (size_t)(m0 + j + (lhi << 3));
      out[m * D_ + ni * 16 + lrow] = c[j] + bias;
    }
  }
}

extern "C" void kernel_launch(void* const* d_in, const int* in_sizes, int n_in,
                              void* d_out, int out_size, void* d_ws, size_t ws_size,
                              hipStream_t stream) {
  const float* X    = (const float*)d_in[0];  // [B, WN, S, D]
  const float* Wqkv = (const float*)d_in[1];  // [3D, D]
  const float* pe   = (const float*)d_in[2];  // [R, D]
  const float* Wo   = (const float*)d_in[3];  // [D, D]
  const float* bo   = (const float*)d_in[4];  // [D]
  float* out = (float*)d_out;                 // [B, WN, S, D]
  float* att = (float*)d_ws;                  // [BW, S, D] = 147 MB scratch

  winmsa_attn_kernel<<<BW_ * H_, 256, 0, stream>>>(X, Wqkv, pe, att);
  winmsa_proj_kernel<<<(BW_ * S_) / K2_ROWS, 256, 0, stream>>>(att, Wo, bo, out);
}